// SPR_RGCN_88648124990354
// MI455X (gfx1250) — compile-verified
//
#include <hip/hip_runtime.h>

// R-GCN on MI455X (gfx1250, wave32).
// Bandwidth-bound in the edge scatter (~1.6 GB/layer at 23.3 TB/s); the fused
// einsum+root GEMM (N x 1152 x 128, ~15 GFLOP/layer) is tiny, so we keep full
// f32 precision and use V_WMMA_F32_16X16X4_F32 for the matrix math.
// A-tiles are staged into LDS with GLOBAL_LOAD_ASYNC_TO_LDS_B128 (ASYNCcnt).

#define RR   8
#define DH   128                 // D_IN == HID
#define KTOT (RR * DH + DH)      // 1152 : [8 relations * 128 | root 128]
#define KC   64                  // K-chunk staged in LDS (divides 1024 & 128)

typedef __attribute__((ext_vector_type(2))) float v2f;
typedef __attribute__((ext_vector_type(8))) float v8f;

// ---------------------------------------------------------------- zero fill
__global__ void k_zero(float* __restrict__ p, long long n) {
  long long i = (long long)blockIdx.x * blockDim.x + threadIdx.x;
  long long stride = (long long)gridDim.x * blockDim.x;
  for (; i < n; i += stride) p[i] = 0.0f;
}

// ------------------------------------------------- per-edge segment scatter
// One wave per edge: 32 lanes x float4 = 128 features.
__global__ void k_scatter(const float* __restrict__ x, const int* __restrict__ ei,
                          const int* __restrict__ et, float* __restrict__ msg,
                          float* __restrict__ cnt, int E) {
  int wave = blockIdx.x * (blockDim.x >> 5) + (threadIdx.x >> 5);
  int lane = threadIdx.x & 31;
  if (wave >= E) return;                       // uniform per wave
  int src = ei[wave];
  int dst = ei[E + wave];
  long long seg = (long long)dst * RR + et[wave];
  const float4 v = *(const float4*)(x + (long long)src * DH + lane * 4);
  float* m = msg + seg * DH + lane * 4;
  unsafeAtomicAdd(m + 0, v.x);
  unsafeAtomicAdd(m + 1, v.y);
  unsafeAtomicAdd(m + 2, v.z);
  unsafeAtomicAdd(m + 3, v.w);
  if (lane == 0) unsafeAtomicAdd(cnt + seg, 1.0f);
}

// --------------------------------------------- mean: msg[seg] /= max(cnt,1)
__global__ void k_scale(float* __restrict__ msg, const float* __restrict__ cnt,
                        long long nseg) {
  long long seg = blockIdx.x;
  if (seg >= nseg) return;
  float inv = 1.0f / fmaxf(cnt[seg], 1.0f);
  msg[seg * DH + threadIdx.x] *= inv;
}

// ---------------------------------------------------------- fused WMMA GEMM
// out[N,128] = relu( [meanScaled | xin] @ [W ; root] + bias )
// Block: 256 thr = 8 waves; block owns 16 output rows, wave w owns cols
// [16w,16w+16). A chunk (16 x KC) staged in LDS (stride 68: conflict-free,
// 16B-aligned rows) via per-lane async b128 copies. K loop: 288 WMMAs.
__global__ void __launch_bounds__(256) k_rgcn_gemm(
    const float* __restrict__ msgS,   // [N, R*DH] already mean-scaled
    const float* __restrict__ xin,    // [N, DH]
    const float* __restrict__ W,      // [R*DH, DH] row-major (== (R,D,H))
    const float* __restrict__ root,   // [DH, DH]
    const float* __restrict__ bias,   // [DH]
    float* __restrict__ out,          // [N, DH]
    int nrows) {
  __shared__ __align__(16) float As[16 * 68];

  const int m0    = blockIdx.x * 16;
  const int tid   = threadIdx.x;
  const int wave  = tid >> 5;                 // 0..7 -> N-tile
  const int lane  = tid & 31;
  const int mr    = lane & 15;                // A-fragment row
  const int khalf = (lane >> 4) * 2;          // ISA: lanes16-31 hold K=+2,+3
  const int col   = wave * 16 + (lane & 15);  // output/B column

  // cooperative A loader: thread -> (row, 4-col group)
  const int lrow = tid >> 4;                  // 0..15
  const int lcg  = (tid & 15) * 4;            // 0..60
  int grow = m0 + lrow;
  if (grow >= nrows) grow = nrows - 1;        // clamp (stores are guarded)

  // LDS byte offset for this thread's 16B slot (low 32 bits of the generic
  // address of a __shared__ object == LDS aperture offset).
  const unsigned lds_off = (unsigned)(size_t)(&As[lrow * 68 + lcg]);

  v8f acc = {};

  for (int k0 = 0; k0 < KTOT; k0 += KC) {
    // KC divides both the relation block (1024) and root block (128), so each
    // chunk comes entirely from msgS or entirely from xin.
    const float* asrc = (k0 < RR * DH)
        ? (msgS + (long long)grow * (RR * DH) + k0 + lcg)
        : (xin + (long long)grow * DH + (k0 - RR * DH) + lcg);
    // async copy 16B/lane: global -> LDS, no VGPR round-trip (ASYNCcnt)
    asm volatile("global_load_async_to_lds_b128 %0, %1, off"
                 :: "v"(lds_off), "v"(asrc) : "memory");
    asm volatile("s_wait_asynccnt 0x0" ::: "memory");
    __syncthreads();

    const float* Bsrc = (k0 < RR * DH)
        ? (W + (long long)k0 * DH)
        : (root + (long long)(k0 - RR * DH) * DH);

#pragma unroll
    for (int ks = 0; ks < KC / 4; ++ks) {
      const int kk = ks * 4 + khalf;
      v2f a, b;
      a.x = As[mr * 68 + kk];
      a.y = As[mr * 68 + kk + 1];
      b.x = Bsrc[(long long)kk * DH + col];
      b.y = Bsrc[(long long)(kk + 1) * DH + col];
      acc = __builtin_amdgcn_wmma_f32_16x16x4_f32(
          false, a, false, b, (short)0, acc, false, false);
    }
    __syncthreads();
  }

  // D layout: VGPR j -> row j (lanes 0-15) / row j+8 (lanes 16-31)
  const float bv = bias[col];
  const int mbase = m0 + ((lane >> 4) * 8);
#pragma unroll
  for (int j = 0; j < 8; ++j) {
    int row = mbase + j;
    if (row < nrows)
      out[(long long)row * DH + col] = fmaxf(acc[j] + bv, 0.0f);
  }
}

// ------------------------------------------------------- global mean pool
__global__ void k_pool(const float* __restrict__ h, const int* __restrict__ batch,
                       float* __restrict__ gsum, float* __restrict__ gcnt, int N) {
  long long i = (long long)blockIdx.x * blockDim.x + threadIdx.x;
  long long total = (long long)N * DH;
  long long stride = (long long)gridDim.x * blockDim.x;
  for (; i < total; i += stride) {
    int n  = (int)(i >> 7);
    int hh = (int)(i & (DH - 1));
    int g  = batch[n];
    unsafeAtomicAdd(gsum + (long long)g * DH + hh, h[i]);
    if (hh == 0) unsafeAtomicAdd(gcnt + g, 1.0f);
  }
}

// ---------------------------------------------------------- classifier head
__global__ void k_head(const float* __restrict__ gsum, const float* __restrict__ gcnt,
                       const float* __restrict__ lw, const float* __restrict__ lb,
                       float* __restrict__ out, int CLS) {
  int g = blockIdx.x;
  int c = threadIdx.x;
  if (c >= CLS) return;
  float inv = 1.0f / fmaxf(gcnt[g], 1.0f);
  float s = 0.0f;
  for (int hh = 0; hh < DH; ++hh)
    s += gsum[(long long)g * DH + hh] * inv * lw[hh * CLS + c];
  out[g * CLS + c] = s + lb[c];
}

extern "C" void kernel_launch(void* const* d_in, const int* in_sizes, int n_in,
                              void* d_out, int out_size, void* d_ws, size_t ws_size,
                              hipStream_t stream) {
  const float* x     = (const float*)d_in[0];
  const int*   ei    = (const int*)d_in[1];
  const int*   et    = (const int*)d_in[2];
  const int*   batch = (const int*)d_in[3];
  const float* W1    = (const float*)d_in[4];
  const float* root1 = (const float*)d_in[5];
  const float* b1    = (const float*)d_in[6];
  const float* W2    = (const float*)d_in[7];
  const float* root2 = (const float*)d_in[8];
  const float* b2    = (const float*)d_in[9];
  const float* lw    = (const float*)d_in[10];
  const float* lb    = (const float*)d_in[11];
  float* out = (float*)d_out;

  const int N   = in_sizes[3];          // batch has N elements
  const int E   = in_sizes[2];          // edge_type has E elements
  const int CLS = in_sizes[11];         // lin_b
  const int G   = out_size / CLS;

  // workspace layout (floats): msg | cnt | h1 | h2 | gsum | gcnt  (~260 MB)
  float* msg  = (float*)d_ws;
  float* cnt  = msg  + (long long)N * RR * DH;
  float* h1   = cnt  + (long long)N * RR;
  float* h2   = h1   + (long long)N * DH;
  float* gsum = h2   + (long long)N * DH;
  float* gcnt = gsum + (long long)G * DH;

  const long long msgcnt_n = (long long)N * RR * DH + (long long)N * RR;
  const int mtiles = (N + 15) / 16;
  const int sblocks = (E + 7) / 8;      // 8 waves (edges) per 256-thread block

  // ---- layer 1 ----
  k_zero<<<2048, 256, 0, stream>>>(msg, msgcnt_n);
  k_scatter<<<sblocks, 256, 0, stream>>>(x, ei, et, msg, cnt, E);
  k_scale<<<N * RR, DH, 0, stream>>>(msg, cnt, (long long)N * RR);
  k_rgcn_gemm<<<mtiles, 256, 0, stream>>>(msg, x, W1, root1, b1, h1, N);

  // ---- layer 2 ----
  k_zero<<<2048, 256, 0, stream>>>(msg, msgcnt_n);
  k_scatter<<<sblocks, 256, 0, stream>>>(h1, ei, et, msg, cnt, E);
  k_scale<<<N * RR, DH, 0, stream>>>(msg, cnt, (long long)N * RR);
  k_rgcn_gemm<<<mtiles, 256, 0, stream>>>(msg, h1, W2, root2, b2, h2, N);

  // ---- pool + head ----
  k_zero<<<64, 256, 0, stream>>>(gsum, (long long)G * DH + G);
  k_pool<<<4096, 256, 0, stream>>>(h2, batch, gsum, gcnt, N);
  k_head<<<G, 32, 0, stream>>>(gsum, gcnt, lw, lb, out, CLS);
}